// MLP_79611513799302
// MI455X (gfx1250) — compile-verified
//
#include <hip/hip_runtime.h>
#include <hip/hip_bf16.h>
#include <math.h>

// Problem sizes (fixed by the reference)
#define SEQ    512
#define DIN    1024
#define HID    512

typedef float v2f __attribute__((ext_vector_type(2)));
typedef float v8f __attribute__((ext_vector_type(8)));

// ---------------------------------------------------------------------------
// Phase 2 (placed first so the disasm snippet shows its inner loop):
// scores[i,j] = sum_d w2[d] * tanh(H[i,d] + M[j,d]) + b2
// 256 threads / block -> one 16x16 output tile. H/M rows staged through LDS
// in 256-wide chunks; pitch 260 floats => bank-stride 4 for the M[j,:] read
// pattern (conflict-free across the 16 distinct j rows in a wave).
// w2 (512 floats) is loaded into LDS once up front.
// ---------------------------------------------------------------------------
__device__ __forceinline__ float hw_tanh(float x)
{
#if __has_builtin(__builtin_amdgcn_tanhf)
    return __builtin_amdgcn_tanhf(x);
#else
    return tanhf(x);
#endif
}

#define CHUNK 256
#define PITCH 260

__global__ void pair_tanh_score(const float* __restrict__ Hm,
                                const float* __restrict__ Mm,
                                const float* __restrict__ w2,
                                const float* __restrict__ b2p,
                                float* __restrict__ out)
{
    __shared__ float sH[16 * PITCH];
    __shared__ float sM[16 * PITCH];
    __shared__ float sW[HID];

    const int t  = threadIdx.x;           // 0..255
    const int ti = t >> 4;                // output row within tile (i)
    const int tj = t & 15;                // output col within tile (j)
    const int i0 = blockIdx.y * 16;
    const int j0 = blockIdx.x * 16;

    // Full w2 into LDS once: 512 floats / 256 threads = 2 each.
    sW[t]       = w2[t];
    sW[t + 256] = w2[t + 256];

    float acc = 0.0f;

    for (int d0 = 0; d0 < HID; d0 += CHUNK) {
        // Cooperative tile load: 16 rows x 256 floats = 1024 float4 per tile,
        // 256 threads -> 4 float4 each per tile. Fully coalesced.
        int idx = t;
#pragma unroll
        for (int rep = 0; rep < 4; ++rep, idx += 256) {
            const int r  = idx >> 6;         // 64 float4 per row
            const int c4 = (idx & 63) * 4;
            const float4 hv = *(const float4*)(Hm + (size_t)(i0 + r) * HID + d0 + c4);
            const float4 mv = *(const float4*)(Mm + (size_t)(j0 + r) * HID + d0 + c4);
            *(float4*)(&sH[r * PITCH + c4]) = hv;
            *(float4*)(&sM[r * PITCH + c4]) = mv;
        }
        __syncthreads();

        const float* __restrict__ hrow = &sH[ti * PITCH];
        const float* __restrict__ mrow = &sM[tj * PITCH];
        const float* __restrict__ wrow = &sW[d0];
#pragma unroll 8
        for (int d = 0; d < CHUNK; ++d) {
            const float v = hrow[d] + mrow[d];
            acc = fmaf(wrow[d], hw_tanh(v), acc);
        }
        __syncthreads();
    }

    out[(size_t)(i0 + ti) * SEQ + (j0 + tj)] = acc + b2p[0];
}

// ---------------------------------------------------------------------------
// Phase 1: Out[s,h] = sum_d X[s,d] * W[h,d] + bias[h]
// One wave32 computes one 16x16 tile of Out using V_WMMA_F32_16X16X4_F32.
//
// A (16x4 f32) per-lane layout (ISA 7.12.2):
//   lanes 0-15 : v0 = A[M=lane, K=0], v1 = A[M=lane, K=1]
//   lanes16-31 : v0 = A[M=lane-16, K=2], v1 = A[M=lane-16, K=3]
// B (4x16 f32) mirrors with N in the lane index; since B[k,n] = W[n,k],
// each lane loads 2 consecutive floats of a W row.
// C/D (16x16 f32, 8 VGPRs): lane<16 -> N=lane, rows M=r (v[r]); lane>=16 ->
//   N=lane-16, rows M=8+r.
// ---------------------------------------------------------------------------
__global__ void gemm_bias_wmma_f32(const float* __restrict__ X,
                                   const float* __restrict__ W,
                                   const float* __restrict__ bias,
                                   float* __restrict__ Out)
{
    const int wavesPerBlock = blockDim.x >> 5;
    const int wave = blockIdx.x * wavesPerBlock + (threadIdx.x >> 5);
    const int lane = threadIdx.x & 31;

    const int tilesN = HID / 16;          // 32 tiles along h
    const int tm = wave / tilesN;         // s-tile
    const int tn = wave % tilesN;         // h-tile
    const int s0 = tm * 16;
    const int h0 = tn * 16;

    const int half = lane >> 4;           // 0: K={0,1}, 1: K={2,3}
    const int l    = lane & 15;

    const float* __restrict__ arow = X + (size_t)(s0 + l) * DIN + 2 * half;
    const float* __restrict__ brow = W + (size_t)(h0 + l) * DIN + 2 * half;

    v8f acc = {};
#pragma unroll 8
    for (int k0 = 0; k0 < DIN; k0 += 4) {
        v2f a = *(const v2f*)(arow + k0);
        v2f b = *(const v2f*)(brow + k0);
        acc = __builtin_amdgcn_wmma_f32_16x16x4_f32(
            /*neg_a=*/false, a, /*neg_b=*/false, b,
            /*c_mod=*/(short)0, acc, /*reuse_a=*/false, /*reuse_b=*/false);
    }

    const float bv = bias[h0 + l];
#pragma unroll
    for (int r = 0; r < 8; ++r) {
        Out[(size_t)(s0 + r + 8 * half) * HID + h0 + l] = acc[r] + bv;
    }
}

// ---------------------------------------------------------------------------
// Launch: two WMMA GEMMs into workspace, then the pairwise tanh reduction.
// Workspace use: H [512*512 f32] + M [512*512 f32] = 2 MB.
// ---------------------------------------------------------------------------
extern "C" void kernel_launch(void* const* d_in, const int* in_sizes, int n_in,
                              void* d_out, int out_size, void* d_ws, size_t ws_size,
                              hipStream_t stream)
{
    const float* x   = (const float*)d_in[0];  // [1, 512, 1024]
    const float* W_h = (const float*)d_in[1];  // [512, 1024]
    const float* b_h = (const float*)d_in[2];  // [512]
    const float* W_m = (const float*)d_in[3];  // [512, 1024]
    const float* b_m = (const float*)d_in[4];  // [512]
    const float* w2  = (const float*)d_in[5];  // [512]
    const float* b2  = (const float*)d_in[6];  // scalar (1 elem)
    float* out = (float*)d_out;                // [512, 512]

    float* Hbuf = (float*)d_ws;
    float* Mbuf = Hbuf + (size_t)SEQ * HID;

    // 32x32 = 1024 tiles per matrix; 4 waves (128 threads) per block -> 256 blocks.
    const int tiles = (SEQ / 16) * (HID / 16);
    const int blk   = 128;
    const int grid  = tiles / (blk / 32);
    gemm_bias_wmma_f32<<<grid, blk, 0, stream>>>(x, W_h, b_h, Hbuf);
    gemm_bias_wmma_f32<<<grid, blk, 0, stream>>>(x, W_m, b_m, Mbuf);

    // 16x16 output tile per 256-thread block; j along x for coalesced stores.
    dim3 g2(SEQ / 16, SEQ / 16);
    pair_tanh_score<<<g2, 256, 0, stream>>>(Hbuf, Mbuf, w2, b2, out);
}